// NGCFModel_35493609734828
// MI455X (gfx1250) — compile-verified
//
#include <hip/hip_runtime.h>
#include <hip/hip_bf16.h>

typedef __attribute__((ext_vector_type(2))) float    v2f;
typedef __attribute__((ext_vector_type(8))) float    v8f;
typedef __attribute__((ext_vector_type(4))) unsigned v4u;
typedef __attribute__((ext_vector_type(8))) unsigned v8u;

#define NUM_USERS 25000
#define NUM_ITEMS 25000
#define N_NODES   50000
#define N_EDGES   800000
#define K_EMB     64
#define ALPHA     0.2f
#define EPSV      1e-12f

// ---------------------------------------------------------------------------
// init: ego = concat(Gu0, Gi0); also write slot 0 (raw ego) into gu/gi output
// ---------------------------------------------------------------------------
__global__ void ngcf_init_kernel(const float* __restrict__ Gu0,
                                 const float* __restrict__ Gi0,
                                 float* __restrict__ ego,
                                 float* __restrict__ out_gu,
                                 float* __restrict__ out_gi) {
    int idx = blockIdx.x * blockDim.x + threadIdx.x;
    if (idx >= N_NODES * K_EMB) return;
    int node = idx >> 6;
    int col  = idx & 63;
    float v;
    if (node < NUM_USERS) {
        v = Gu0[idx];
        out_gu[(size_t)node * 256 + col] = v;
    } else {
        v = Gi0[idx - NUM_USERS * K_EMB];
        out_gi[(size_t)(node - NUM_USERS) * 256 + col] = v;
    }
    ego[idx] = v;
}

__global__ void ngcf_zero_kernel(float* __restrict__ p, int n) {
    int i = blockIdx.x * blockDim.x + threadIdx.x;
    if (i < n) p[i] = 0.0f;
}

// ---------------------------------------------------------------------------
// SpMM scatter: side[r] += v * ego[c] for each edge; one wave32 per edge,
// each lane handles 2 of the 64 columns.
// ---------------------------------------------------------------------------
__global__ void ngcf_spmm_kernel(const int* __restrict__ rows,
                                 const int* __restrict__ cols,
                                 const float* __restrict__ vals,
                                 const float* __restrict__ ego,
                                 float* __restrict__ side) {
    int wave = (blockIdx.x * blockDim.x + threadIdx.x) >> 5;
    int lane = threadIdx.x & 31;
    if (wave >= N_EDGES) return;
    int   r = rows[wave];
    int   c = cols[wave];
    float v = vals[wave];
    const float* src = ego + (size_t)c * K_EMB + lane * 2;
    float x0 = src[0];
    float x1 = src[1];
    float* dst = side + (size_t)r * K_EMB + lane * 2;
    atomicAdd(dst + 0, v * x0);
    atomicAdd(dst + 1, v * x1);
}

// ---------------------------------------------------------------------------
// TDM: DMA a dense 64x64 f32 tile (16 KB) from global into LDS.
// D# group0: count=1, lds_addr, global_addr[56:0], type=2 ("image").
// D# group1: wg_mask=0, data_size=2 (4B), tensor_dim0/1=64, tile_dim0/1=64,
//            tensor_dim0_stride=64. 2-operand form => groups 2/3 NULL (2D).
// Tracked by TENSORcnt; wave issuing it must s_wait_tensorcnt before use.
// ---------------------------------------------------------------------------
__device__ __forceinline__ void tdm_load_w64x64(const float* gptr,
                                                unsigned lds_off) {
    unsigned long long ga = (unsigned long long)(uintptr_t)gptr;
    v4u g0;
    g0[0] = 1u;                                            // count=1, user mode
    g0[1] = lds_off;                                       // lds_addr (bytes)
    g0[2] = (unsigned)(ga & 0xFFFFFFFFull);                // global_addr[31:0]
    g0[3] = (unsigned)((ga >> 32) & 0x01FFFFFFull)         // global_addr[56:32]
            | (2u << 30);                                  // type=2
    v8u g1;
    g1[0] = 0x00020000u;      // workgroup_mask=0, data_size=2 (4 bytes)
    g1[1] = 64u << 16;        // tensor_dim0[15:0] = 64 at bits [63:48]
    g1[2] = 64u << 16;        // tensor_dim1[15:0] = 64 at bits [95:80]
    g1[3] = 64u << 16;        // tile_dim0 = 64 at bits [127:112]
    g1[4] = 64u;              // tile_dim1 = 64 at bits [143:128]
    g1[5] = 64u;              // tensor_dim0_stride = 64 at bits [191:160]
    g1[6] = 0u;
    g1[7] = 0u;
    asm volatile("tensor_load_to_lds %0, %1" :: "s"(g0), "s"(g1) : "memory");
}

// ---------------------------------------------------------------------------
// Fused layer: D1 = side@Wgc + bgc (leaky), D2 = (ego*side)@Wbi + bbi (leaky),
// ego_new = D1+D2 (written back un-normalized), norm(ego_new) -> output slot.
// 128 threads = 4 waves; block handles 16 rows x 64 cols; wave w owns cols
// [16w,16w+16). Weights staged to LDS via TDM (overlapped with tile staging);
// f32 WMMA 16x16x4, K accumulated over 16 steps.
// ---------------------------------------------------------------------------
__global__ __launch_bounds__(128)
void ngcf_layer_kernel(const float* __restrict__ side,
                       float* __restrict__ ego,
                       const float* __restrict__ Wgc,
                       const float* __restrict__ bgc,
                       const float* __restrict__ Wbi,
                       const float* __restrict__ bbi,
                       float* __restrict__ out_gu,
                       float* __restrict__ out_gi,
                       int slot) {
    __shared__ float s_wgc[64 * 64];   // TDM destination (row-major, stride 64)
    __shared__ float s_wbi[64 * 64];
    __shared__ float s_side[16][68];   // stride 68: conflict-free A-frag reads
    __shared__ float s_prod[16][68];
    __shared__ float s_ego[16][65];
    __shared__ float s_rnorm[16];

    const int row0 = blockIdx.x * 16;
    const int tid  = threadIdx.x;

    // Wave 0 kicks off the two weight-tile DMAs; they run while all waves
    // stage the side / (ego*side) tiles through VGPRs below.
    if (tid < 32) {
        tdm_load_w64x64(Wgc, (unsigned)(uintptr_t)&s_wgc[0]);
        tdm_load_w64x64(Wbi, (unsigned)(uintptr_t)&s_wbi[0]);
    }

    for (int i = tid; i < 16 * 64; i += 128) {
        int r = i >> 6, c = i & 63;
        float sv = side[(size_t)(row0 + r) * K_EMB + c];
        float ev = ego [(size_t)(row0 + r) * K_EMB + c];
        s_side[r][c] = sv;
        s_prod[r][c] = ev * sv;
    }

    if (tid < 32) __builtin_amdgcn_s_wait_tensorcnt(0);
    __syncthreads();

    const int wave = tid >> 5;
    const int lane = tid & 31;
    const int m    = lane & 15;          // A row
    const int kb   = (lane >> 4) * 2;    // A/B K sub-offset per lane half
    const int col0 = wave * 16;
    const int n    = col0 + (lane & 15); // B/C/D column

    v8f acc1 = {0.f,0.f,0.f,0.f,0.f,0.f,0.f,0.f};
    v8f acc2 = {0.f,0.f,0.f,0.f,0.f,0.f,0.f,0.f};

    for (int k0 = 0; k0 < K_EMB; k0 += 4) {
        v2f a1, a2, b1, b2;
        a1.x = s_side[m][k0 + kb];
        a1.y = s_side[m][k0 + kb + 1];
        a2.x = s_prod[m][k0 + kb];
        a2.y = s_prod[m][k0 + kb + 1];
        b1.x = s_wgc[(k0 + kb)     * 64 + n];
        b1.y = s_wgc[(k0 + kb + 1) * 64 + n];
        b2.x = s_wbi[(k0 + kb)     * 64 + n];
        b2.y = s_wbi[(k0 + kb + 1) * 64 + n];
        acc1 = __builtin_amdgcn_wmma_f32_16x16x4_f32(false, a1, false, b1,
                                                     (short)0, acc1, false, false);
        acc2 = __builtin_amdgcn_wmma_f32_16x16x4_f32(false, a2, false, b2,
                                                     (short)0, acc2, false, false);
    }

    // Epilogue: bias + leaky relu + sum -> LDS ego tile
    float bg = bgc[n];
    float bb = bbi[n];
    #pragma unroll
    for (int j = 0; j < 8; ++j) {
        int r = (lane >> 4) * 8 + j;
        float v1 = acc1[j] + bg;
        v1 = v1 > 0.f ? v1 : ALPHA * v1;
        float v2 = acc2[j] + bb;
        v2 = v2 > 0.f ? v2 : ALPHA * v2;
        s_ego[r][n] = v1 + v2;
    }
    __syncthreads();

    // Per-row L2 norm scale
    if (tid < 16) {
        float s = 0.f;
        #pragma unroll 8
        for (int c = 0; c < 64; ++c) {
            float x = s_ego[tid][c];
            s += x * x;
        }
        s_rnorm[tid] = rsqrtf(fmaxf(s, EPSV));
    }
    __syncthreads();

    // Write back: un-normalized ego (next layer input), normalized to output
    for (int i = tid; i < 16 * 64; i += 128) {
        int r = i >> 6, c = i & 63;
        int node = row0 + r;
        float e  = s_ego[r][c];
        ego[(size_t)node * K_EMB + c] = e;
        float nv = e * s_rnorm[r];
        if (node < NUM_USERS)
            out_gu[(size_t)node * 256 + slot * 64 + c] = nv;
        else
            out_gi[(size_t)(node - NUM_USERS) * 256 + slot * 64 + c] = nv;
    }
}

// ---------------------------------------------------------------------------
// xui[i] = dot(gu[user[i]], gi[item[i]]) over 256 dims; one wave per sample.
// ---------------------------------------------------------------------------
__global__ void ngcf_dot_kernel(const int* __restrict__ user,
                                const int* __restrict__ item,
                                const float* __restrict__ gu,
                                const float* __restrict__ gi,
                                float* __restrict__ xui) {
    int wid  = (blockIdx.x * blockDim.x + threadIdx.x) >> 5;
    int lane = threadIdx.x & 31;
    if (wid >= 4096) return;
    int u  = user[wid];
    int it = item[wid];
    const float* pu = gu + (size_t)u  * 256;
    const float* pi = gi + (size_t)it * 256;
    float s = 0.f;
    #pragma unroll
    for (int j = 0; j < 8; ++j)
        s += pu[lane + 32 * j] * pi[lane + 32 * j];
    #pragma unroll
    for (int off = 16; off > 0; off >>= 1)
        s += __shfl_xor(s, off, 32);
    if (lane == 0) xui[wid] = s;
}

extern "C" void kernel_launch(void* const* d_in, const int* in_sizes, int n_in,
                              void* d_out, int out_size, void* d_ws, size_t ws_size,
                              hipStream_t stream) {
    const int*   adj_rows = (const int*)  d_in[0];
    const int*   adj_cols = (const int*)  d_in[1];
    const float* adj_vals = (const float*)d_in[2];
    const float* Gu0      = (const float*)d_in[3];
    const float* Gi0      = (const float*)d_in[4];
    const float* W_gc     = (const float*)d_in[5];
    const float* b_gc     = (const float*)d_in[6];
    const float* W_bi     = (const float*)d_in[7];
    const float* b_bi     = (const float*)d_in[8];
    const int*   user     = (const int*)  d_in[9];
    const int*   item     = (const int*)  d_in[10];

    float* out = (float*)d_out;
    float* xui = out;                                   // [4096]
    float* gu  = out + 4096;                            // [25000 x 256]
    float* gi  = gu + (size_t)NUM_USERS * 256;          // [25000 x 256]

    float* ego  = (float*)d_ws;                         // [50000 x 64]
    float* side = ego + (size_t)N_NODES * K_EMB;        // [50000 x 64]

    const int NE = N_NODES * K_EMB;

    ngcf_init_kernel<<<(NE + 255) / 256, 256, 0, stream>>>(Gu0, Gi0, ego, gu, gi);

    for (int k = 0; k < 3; ++k) {
        ngcf_zero_kernel<<<(NE + 255) / 256, 256, 0, stream>>>(side, NE);
        ngcf_spmm_kernel<<<(N_EDGES + 7) / 8, 256, 0, stream>>>(
            adj_rows, adj_cols, adj_vals, ego, side);
        ngcf_layer_kernel<<<N_NODES / 16, 128, 0, stream>>>(
            side, ego,
            W_gc + (size_t)k * K_EMB * K_EMB, b_gc + (size_t)k * K_EMB,
            W_bi + (size_t)k * K_EMB * K_EMB, b_bi + (size_t)k * K_EMB,
            gu, gi, k + 1);
    }

    ngcf_dot_kernel<<<(4096 * 32) / 256, 256, 0, stream>>>(user, item, gu, gi, xui);
}